// CR_72773925863437
// MI455X (gfx1250) — compile-verified
//
#include <hip/hip_runtime.h>
#include <stdint.h>

// CR gate, D=2, control=qudit0, target=qudit1, L=24.
// out = U x, U block-diagonal: control=0 -> identity (pure copy),
// control=1 -> [[c, -i s],[-i s, c]] on target axis, c=cos(a/2), s=sin(a/2).
//
// Memory-bound: 256 MB traffic -> ~11 us floor at 23.3 TB/s.
// Single fused kernel; region split is uniform per block (no divergence).

typedef float v4f __attribute__((ext_vector_type(4)));

#define N_AMP        (1u << 24)   // 2^24 amplitudes
#define REST         (1u << 22)   // trailing-dim block size
#define HALF         (1u << 23)   // start of control=1 half
#define COPY_BLOCKS  ((HALF / 4u) / 256u)   // 8192
#define MIX_BLOCKS   ((REST / 4u) / 256u)   // 4096

__global__ __launch_bounds__(256) void cr_fused(
    const float* __restrict__ xr, const float* __restrict__ xi,
    const float* __restrict__ angle,
    float* __restrict__ outr, float* __restrict__ outi)
{
    __shared__ float smem[256 * 8];     // 2 x b128 per lane = 8 KB (copy path)

    if (blockIdx.x < COPY_BLOCKS) {
        // ------------------------------------------------------------------
        // Control = 0 half: pure copy of 2^23 floats from each plane.
        // CDNA5 async global<->LDS DMA path (ASYNCcnt): memory -> LDS ->
        // memory without touching VGPRs. One b128 per plane per thread.
        // ------------------------------------------------------------------
        const unsigned t = blockIdx.x * 256u + threadIdx.x;
        const unsigned i = t * 4u;

        // Low 32 bits of a flat shared-memory address == workgroup-relative
        // LDS byte offset (flat LDS addr = {SHARED_BASE, offset[31:0]}).
        unsigned lds0 = (unsigned)(uintptr_t)(&smem[threadIdx.x * 8u]);
        unsigned lds1 = lds0 + 16u;

        const float* srcR = xr + i;
        const float* srcI = xi + i;
        float* dstR = outr + i;
        float* dstI = outi + i;

        asm volatile(
            "global_load_async_to_lds_b128 %0, %2, off\n\t"
            "global_load_async_to_lds_b128 %1, %3, off\n\t"
            "s_wait_asynccnt 0\n\t"
            "global_store_async_from_lds_b128 %4, %0, off\n\t"
            "global_store_async_from_lds_b128 %5, %1, off\n\t"
            "s_wait_asynccnt 0\n\t"
            :
            : "v"(lds0), "v"(lds1), "v"(srcR), "v"(srcI), "v"(dstR), "v"(dstI)
            : "memory");
    } else {
        // ------------------------------------------------------------------
        // Control = 1 half: 2x2 complex rotation between target=0 block
        // [HALF, HALF+REST) and target=1 block [HALF+REST, 2*HALF).
        // One thread handles 4 consecutive elements of BOTH blocks:
        // 4 NT b128 loads, 16 FMAs, 4 NT b128 stores.
        // ------------------------------------------------------------------
        const unsigned u = (blockIdx.x - COPY_BLOCKS) * 256u + threadIdx.x;
        const unsigned i = HALF + u * 4u;   // control=1, target=0
        const unsigned p = i + REST;        // control=1, target=1

        float s, c;
        sincosf(0.5f * angle[0], &s, &c);   // uniform scalar; once per thread

        v4f xr0 = __builtin_nontemporal_load((const v4f*)(xr + i));
        v4f xi0 = __builtin_nontemporal_load((const v4f*)(xi + i));
        v4f xr1 = __builtin_nontemporal_load((const v4f*)(xr + p));
        v4f xi1 = __builtin_nontemporal_load((const v4f*)(xi + p));

        v4f or0 = c * xr0 + s * xi1;        // re(c*x10 - i s x11)
        v4f oi0 = c * xi0 - s * xr1;        // im(c*x10 - i s x11)
        v4f or1 = s * xi0 + c * xr1;        // re(-i s x10 + c x11)
        v4f oi1 = c * xi1 - s * xr0;        // im(-i s x10 + c x11)

        __builtin_nontemporal_store(or0, (v4f*)(outr + i));
        __builtin_nontemporal_store(oi0, (v4f*)(outi + i));
        __builtin_nontemporal_store(or1, (v4f*)(outr + p));
        __builtin_nontemporal_store(oi1, (v4f*)(outi + p));
    }
}

extern "C" void kernel_launch(void* const* d_in, const int* in_sizes, int n_in,
                              void* d_out, int out_size, void* d_ws, size_t ws_size,
                              hipStream_t stream) {
    (void)in_sizes; (void)n_in; (void)out_size; (void)d_ws; (void)ws_size;

    const float* xr    = (const float*)d_in[0];
    const float* xi    = (const float*)d_in[1];
    const float* angle = (const float*)d_in[2];

    float* outr = (float*)d_out;          // stacked [real, imag] -> (2, N)
    float* outi = (float*)d_out + N_AMP;

    cr_fused<<<COPY_BLOCKS + MIX_BLOCKS, 256, 0, stream>>>(xr, xi, angle, outr, outi);
}